// GraphTripletGCNLayer_18631568130411
// MI455X (gfx1250) — compile-verified
//
#include <hip/hip_runtime.h>
#include <hip/hip_bf16.h>

typedef __attribute__((ext_vector_type(2))) float v2f;
typedef __attribute__((ext_vector_type(8))) float v8f;

#define H      256   // hidden dim
#define K2     512   // concat dim
#define NB_M    64   // M rows per block
#define KB      32   // K chunk per LDS stage
#define A_STR   36   // padded LDS stride for A tile (floats)
#define B_STR  260   // padded LDS stride for B tile (floats)

// Per-lane async global->LDS 16B copy (CDNA5, tracked with ASYNCcnt).
// ldsAddr is the 32-bit LDS byte address (low 32 bits of a generic pointer
// to __shared__ memory per the aperture scheme).
__device__ __forceinline__ void async_g2l_b128(unsigned ldsAddr, const void* gptr) {
    asm volatile("global_load_async_to_lds_b128 %0, %1, off"
                 :: "v"(ldsAddr), "v"(gptr)
                 : "memory");
}
__device__ __forceinline__ void wait_asynccnt0() {
    asm volatile("s_wait_asynccnt 0x0" ::: "memory");
}

// ---------------------------------------------------------------------------
// Kernel 1: zero the aggregation buffer (float4 stores)
// ---------------------------------------------------------------------------
__global__ __launch_bounds__(256) void zero_agg(float* __restrict__ agg, int n4) {
    int i = blockIdx.x * blockDim.x + threadIdx.x;
    if (i < n4) ((float4*)agg)[i] = make_float4(0.f, 0.f, 0.f, 0.f);
}

// ---------------------------------------------------------------------------
// Kernel 2: gather rel/subj/obj rows, scatter-add both messages with HW fp32
// atomics. One thread per (triple, 4-feature quad): 300000 * 64 threads.
// ---------------------------------------------------------------------------
__global__ __launch_bounds__(256) void scatter_msgs(
    const float* __restrict__ node, const float* __restrict__ rel,
    const long long* __restrict__ triples, float* __restrict__ agg,
    int n_triples) {
    int tid = blockIdx.x * blockDim.x + threadIdx.x;
    int t = tid >> 6;            // triple index
    int c = (tid & 63) << 2;     // feature quad offset
    if (t >= n_triples) return;

    long long s = triples[3LL * t + 0];
    long long r = triples[3LL * t + 1];
    long long o = triples[3LL * t + 2];

    const float4 rv = *(const float4*)(rel  + r * (long long)H + c);
    const float4 sv = *(const float4*)(node + s * (long long)H + c);
    const float4 ov = *(const float4*)(node + o * (long long)H + c);

    float* aggO = agg + o * (long long)H + c;   // agg[obj]  += node[subj] + rel
    float* aggS = agg + s * (long long)H + c;   // agg[subj] += node[obj]  + rel

    unsafeAtomicAdd(aggO + 0, sv.x + rv.x);
    unsafeAtomicAdd(aggO + 1, sv.y + rv.y);
    unsafeAtomicAdd(aggO + 2, sv.z + rv.z);
    unsafeAtomicAdd(aggO + 3, sv.w + rv.w);
    unsafeAtomicAdd(aggS + 0, ov.x + rv.x);
    unsafeAtomicAdd(aggS + 1, ov.y + rv.y);
    unsafeAtomicAdd(aggS + 2, ov.z + rv.z);
    unsafeAtomicAdd(aggS + 3, ov.w + rv.w);
}

// ---------------------------------------------------------------------------
// Kernel 3: out = node + silu(concat(node, agg) @ W + b)
// Block: 256 threads = 8 wave32s; tile M=64, N=256 (full width), K chunk 32.
// LDS tiles are filled with GLOBAL_LOAD_ASYNC_TO_LDS_B128 (ASYNCcnt-tracked
// direct memory->LDS DMA, no VGPR staging, single wait per tile).
// Wave w owns C tiles (mt = 0..3) x (nt = 2w, 2w+1) as v_wmma_f32_16x16x4_f32
// accumulators.
// ---------------------------------------------------------------------------
__global__ __launch_bounds__(256) void gemm_silu(
    const float* __restrict__ node, const float* __restrict__ agg,
    const float* __restrict__ W, const float* __restrict__ bias,
    float* __restrict__ out, int n_rows) {

    __shared__ float As[NB_M * A_STR];   // 64 x 32 (+pad)   A tile
    __shared__ float Bs[KB * B_STR];     // 32 x 256 (+pad)  W tile

    const int tid  = threadIdx.x;
    const int wave = tid >> 5;           // 0..7
    const int lane = tid & 31;
    const int half = lane >> 4;          // 0: K+0/K+1, 1: K+2/K+3 (A/B frag halves)
    const int l15  = lane & 15;
    const int row0 = blockIdx.x * NB_M;

    // LDS byte addresses for the async copies this thread performs.
    const int aIdx0 = tid;               // float4 slot 0..255
    const int aIdx1 = tid + 256;         // float4 slot 256..511
    const int aRow0 = aIdx0 >> 3, aCol0 = (aIdx0 & 7) << 2;
    const int aRow1 = aIdx1 >> 3, aCol1 = (aIdx1 & 7) << 2;
    const unsigned aLds0 = (unsigned)(size_t)&As[aRow0 * A_STR + aCol0];
    const unsigned aLds1 = (unsigned)(size_t)&As[aRow1 * A_STR + aCol1];

    v8f acc[4][2];
#pragma unroll
    for (int mt = 0; mt < 4; ++mt)
#pragma unroll
        for (int nt = 0; nt < 2; ++nt)
            acc[mt][nt] = (v8f){0.f, 0.f, 0.f, 0.f, 0.f, 0.f, 0.f, 0.f};

    for (int k0 = 0; k0 < K2; k0 += KB) {
        // ---- async-stage A tile: rows from node (k<256) or agg (k>=256) ----
        const float* Asrc = (k0 < H) ? node : agg;
        const int    ksrc = (k0 < H) ? k0 : (k0 - H);

        int gr0 = row0 + aRow0; if (gr0 >= n_rows) gr0 = n_rows - 1;
        int gr1 = row0 + aRow1; if (gr1 >= n_rows) gr1 = n_rows - 1;
        async_g2l_b128(aLds0, Asrc + (long long)gr0 * H + ksrc + aCol0);
        async_g2l_b128(aLds1, Asrc + (long long)gr1 * H + ksrc + aCol1);

        // ---- async-stage B tile: W rows k0..k0+31, all 256 columns ----
#pragma unroll
        for (int i = 0; i < 8; ++i) {
            int idx = tid + i * 256;          // 0..2047 float4 slots
            int rr  = idx >> 6;               // k row 0..31
            int cc  = (idx & 63) << 2;        // col 0..252
            async_g2l_b128((unsigned)(size_t)&Bs[rr * B_STR + cc],
                           W + (long long)(k0 + rr) * H + cc);
        }
        wait_asynccnt0();
        __syncthreads();

        // ---- 8 WMMA k-steps of 4 over this chunk ----
#pragma unroll
        for (int kk = 0; kk < KB; kk += 4) {
            // A frag (16x4 f32): lane = M row, VGPR0/1 = K even/odd,
            // lane halves select K+0..1 vs K+2..3 -> contiguous 2-float load.
            v2f a[4];
#pragma unroll
            for (int mt = 0; mt < 4; ++mt)
                a[mt] = *(const v2f*)&As[(mt * 16 + l15) * A_STR + kk + 2 * half];

            // B frag (4x16 f32): lane = N col, symmetric K split across halves.
            v2f b[2];
#pragma unroll
            for (int nt = 0; nt < 2; ++nt) {
                int n  = (wave * 2 + nt) * 16 + l15;
                int kb = kk + 2 * half;
                v2f bv;
                bv.x = Bs[kb * B_STR + n];
                bv.y = Bs[(kb + 1) * B_STR + n];
                b[nt] = bv;
            }

#pragma unroll
            for (int mt = 0; mt < 4; ++mt)
#pragma unroll
                for (int nt = 0; nt < 2; ++nt)
                    acc[mt][nt] = __builtin_amdgcn_wmma_f32_16x16x4_f32(
                        /*neg_a=*/false, a[mt], /*neg_b=*/false, b[nt],
                        /*c_mod=*/(short)0, acc[mt][nt],
                        /*reuse_a=*/false, /*reuse_b=*/false);
        }
        __syncthreads();
    }

    // ---- epilogue: bias + SiLU + residual, predicated stores ----
    // C layout: VGPR r, lanes 0-15 -> M=r, N=lane; lanes 16-31 -> M=r+8, N=lane-16
#pragma unroll
    for (int nt = 0; nt < 2; ++nt) {
        int ncol = (wave * 2 + nt) * 16 + l15;
        float bb = bias[ncol];
#pragma unroll
        for (int mt = 0; mt < 4; ++mt) {
#pragma unroll
            for (int r = 0; r < 8; ++r) {
                int grow = row0 + mt * 16 + half * 8 + r;
                if (grow < n_rows) {
                    float x  = acc[mt][nt][r] + bb;
                    float sl = x / (1.0f + __expf(-x));          // silu
                    long long off = (long long)grow * H + ncol;
                    out[off] = node[off] + sl;
                }
            }
        }
    }
}

// ---------------------------------------------------------------------------
extern "C" void kernel_launch(void* const* d_in, const int* in_sizes, int n_in,
                              void* d_out, int out_size, void* d_ws, size_t ws_size,
                              hipStream_t stream) {
    const float*     node    = (const float*)d_in[0];      // (100000, 256) f32
    const float*     rel     = (const float*)d_in[1];      // (100000, 256) f32
    const long long* triples = (const long long*)d_in[2];  // (300000, 3) i64
    const float*     W       = (const float*)d_in[3];      // (512, 256) f32
    const float*     b       = (const float*)d_in[4];      // (256,) f32
    float*           out     = (float*)d_out;
    float*           agg     = (float*)d_ws;               // 100000*256 f32 scratch

    const int n_nodes   = in_sizes[0] / H;   // 100000
    const int n_triples = in_sizes[2] / 3;   // 300000

    // 1) agg = 0
    int n4 = n_nodes * (H / 4);
    zero_agg<<<(n4 + 255) / 256, 256, 0, stream>>>(agg, n4);

    // 2) scatter messages (fp32 HW atomics)
    long long work = (long long)n_triples * 64;
    int sgrid = (int)((work + 255) / 256);
    scatter_msgs<<<sgrid, 256, 0, stream>>>(node, rel, triples, agg, n_triples);

    // 3) fused GEMM + bias + SiLU + residual (fp32 WMMA, async LDS staging)
    int ggrid = (n_nodes + NB_M - 1) / NB_M;
    gemm_silu<<<ggrid, 256, 0, stream>>>(node, agg, W, b, out, n_nodes);
}